// SparseDualFlow_62843961475264
// MI455X (gfx1250) — compile-verified
//
#include <hip/hip_runtime.h>
#include <stdint.h>

// ---- reference hyperparameters ----
#define STEP_SIZE 0.01f
#define MOMENTUM  0.9f
#define N_ITERS   100
#define FLOW_THR  1e-4f

#define BLOCK 256
#define EPT   16                // f32 elements per thread
#define EPV   (EPT / 2)         // float2 pairs per thread
#define TILE  (BLOCK * EPT)     // 4096 floats = 16 KB per block tile

typedef float    float2v __attribute__((ext_vector_type(2)));
typedef unsigned tdm_g0_t __attribute__((ext_vector_type(4)));
typedef unsigned tdm_g1_t __attribute__((ext_vector_type(8)));

// ---------------- packed f32 VOP3P helpers (guaranteed v_pk_*) --------------
__device__ __forceinline__ float2v pk_mul(float2v a, float2v b) {
  float2v d;
  asm("v_pk_mul_f32 %0, %1, %2" : "=v"(d) : "v"(a), "v"(b));
  return d;
}
__device__ __forceinline__ float2v pk_add(float2v a, float2v b) {
  float2v d;
  asm("v_pk_add_f32 %0, %1, %2" : "=v"(d) : "v"(a), "v"(b));
  return d;
}
__device__ __forceinline__ float2v pk_fma(float2v a, float2v b, float2v c) {
  float2v d;
  asm("v_pk_fma_f32 %0, %1, %2, %3" : "=v"(d) : "v"(a), "v"(b), "v"(c));
  return d;
}

// One recurrence step on a packed pair. State kept as (fl, nb=-acc), input as
// ndv=-dual, so every op is a plain pk mul/add/fma (no neg modifiers) and the
// arithmetic is bit-identical to the reference ordering:
//   mb  = 0.9*nb      == -acc_m
//   t   = fl + mb     == fl - acc_m
//   g   = 2*t + ndv   == 2*(fl-acc_m) - dual
//   nb' = -0.01*g+mb  == -(acc_m + 0.01*g)
//   s   = fl + nb'    == fl - acc'
//   fl' = max(s, 0)
__device__ __forceinline__ void sdf_step_pk(float2v ndv, float2v& fl, float2v& nb) {
  const float2v vMOM = {MOMENTUM, MOMENTUM};
  const float2v vTWO = {2.0f, 2.0f};
  const float2v vNST = {-STEP_SIZE, -STEP_SIZE};
  float2v mb = pk_mul(vMOM, nb);
  float2v t  = pk_add(fl, mb);
  float2v g  = pk_fma(vTWO, t, ndv);
  float2v b2 = pk_fma(vNST, g, mb);
  float2v s  = pk_add(fl, b2);
  float2v nf;
  nf.x = fmaxf(s.x, 0.0f);
  nf.y = fmaxf(s.y, 0.0f);
  fl = nf;
  nb = b2;
}

// Same step, also folds |delta flow| of both components into mx.
__device__ __forceinline__ void sdf_step_pk_track(float2v ndv, float2v& fl,
                                                  float2v& nb, float& mx) {
  float2v of = fl;
  sdf_step_pk(ndv, fl, nb);
  float dx = fl.x - of.x;
  float dy = fl.y - of.y;
  mx = fmaxf(mx, fmaxf(__builtin_fabsf(dx), __builtin_fabsf(dy)));  // v_max3 + |mods|
}

// Issue a TDM load of TILE consecutive f32 from gsrc into LDS at lds_byte_off.
// D# per CDNA5 ISA ch.8 (2-group form, 1-row tile of TILE f32 elements).
__device__ __forceinline__ void tdm_load_tile_to_lds(const float* gsrc,
                                                     uint32_t lds_byte_off) {
  uint64_t ga = (uint64_t)(uintptr_t)gsrc;
  tdm_g0_t g0;
  g0.x = 1u;                                    // count=1, user descriptor
  g0.y = lds_byte_off;                          // lds_addr (bytes)
  g0.z = (uint32_t)(ga & 0xFFFFFFFFu);          // global_addr[31:0]
  g0.w = (uint32_t)((ga >> 32) & 0x01FFFFFFu)   // global_addr[56:32]
       | (2u << 30);                            // type = 2 ("image")
  tdm_g1_t g1;
  g1.s0 = (2u << 16);                           // data_size=2 -> 4B; mask=0
  g1.s1 = ((uint32_t)TILE & 0xFFFFu) << 16;     // tensor_dim0[15:0]
  g1.s2 = (((uint32_t)TILE >> 16) & 0xFFFFu)    // tensor_dim0[31:16]
        | (1u << 16);                           // tensor_dim1[15:0] = 1
  g1.s3 = ((uint32_t)TILE & 0xFFFFu) << 16;     // tensor_dim1 hi=0 | tile_dim0
  g1.s4 = 1u;                                   // tile_dim1=1, tile_dim2=0
  g1.s5 = (uint32_t)TILE;                       // tensor_dim0_stride[31:0]
  g1.s6 = 0u;
  g1.s7 = 0u;
  asm volatile("tensor_load_to_lds %0, %1" :: "s"(g0), "s"(g1) : "memory");
}

__device__ __forceinline__ void stage_tile(const float* __restrict__ dual,
                                           float* tile, int tb, int tid, int n) {
  bool full = (tb + TILE <= n);
  if (full) {
    if (tid < 32) {  // wave 0 issues the DMA; TENSORcnt is per-wave
      tdm_load_tile_to_lds(dual + tb, (uint32_t)(uintptr_t)(&tile[0]));
      __builtin_amdgcn_s_wait_tensorcnt(0);
    }
  } else {
    for (int k = 0; k < EPT; ++k) {
      int idx = tb + tid + k * BLOCK;
      tile[tid + k * BLOCK] = (idx < n) ? dual[idx] : 0.0f;
    }
  }
}

// Load negated duals (recurrence only ever uses -dual) and zero the state.
__device__ __forceinline__ void load_state(const float* tile, int tid,
                                           float2v* ndv, float2v* fl, float2v* nb) {
#pragma unroll
  for (int j = 0; j < EPV; ++j) {
    ndv[j].x = -tile[tid + (2 * j + 0) * BLOCK];
    ndv[j].y = -tile[tid + (2 * j + 1) * BLOCK];
    fl[j] = (float2v){0.0f, 0.0f};
    nb[j] = (float2v){0.0f, 0.0f};
  }
}

__device__ __forceinline__ void store_flow(float* __restrict__ out, const float2v* fl,
                                           int tb, int tid, int n) {
#pragma unroll
  for (int j = 0; j < EPV; ++j) {
    int i0 = tb + tid + (2 * j + 0) * BLOCK;
    int i1 = tb + tid + (2 * j + 1) * BLOCK;
    if (i0 < n) out[i0] = fl[j].x;
    if (i1 < n) out[i1] = fl[j].y;
  }
}

// ---------------- Pass 1: 100-bit violation mask + f_100 output -------------
__global__ void __launch_bounds__(BLOCK)
sdf_pass1(const float* __restrict__ dual, unsigned* __restrict__ ws,
          float* __restrict__ out, int n) {
  __shared__ float tile[TILE];
  __shared__ unsigned viol[4];
  const int tb  = blockIdx.x * TILE;
  const int tid = threadIdx.x;

  stage_tile(dual, tile, tb, tid, n);
  if (tid < 4) viol[tid] = 0u;
  __syncthreads();

  float2v ndv[EPV], fl[EPV], nb[EPV];
  load_state(tile, tid, ndv, fl, nb);

  unsigned m[4];
#pragma unroll
  for (int w = 0; w < 4; ++w) {
    unsigned mw = 0u;
    const int nbits = (w < 3) ? 32 : (N_ITERS - 96);  // 32,32,32,4
    for (int b = 0; b < nbits; ++b) {
      float mx = 0.0f;
#pragma unroll
      for (int j = 0; j < EPV; ++j) sdf_step_pk_track(ndv[j], fl[j], nb[j], mx);
      if (mx > FLOW_THR) mw |= (1u << b);
    }
    m[w] = mw;
  }

#pragma unroll
  for (int w = 0; w < 4; ++w) atomicOr(&viol[w], m[w]);  // ds_or
  __syncthreads();
  if (tid < 4) atomicOr(&ws[tid], viol[tid]);            // global_atomic_or

  // After the mask loops the state is exactly f_100: publish it. If the loop
  // never globally converges (the expected case), pass 2 skips entirely.
  store_flow(out, fl, tb, tid, n);
}

// -------- Select iteration count: first step with all-clear, +1; cap 100 ----
__global__ void sdf_select_iters(unsigned* __restrict__ ws) {
  unsigned w0 = ws[0], w1 = ws[1], w2 = ws[2], w3 = ws[3];
  int iters = N_ITERS;
  for (int k = 0; k < N_ITERS; ++k) {
    unsigned word = (k < 32) ? w0 : (k < 64) ? w1 : (k < 96) ? w2 : w3;
    if (((word >> (k & 31)) & 1u) == 0u) { iters = k + 1; break; }
  }
  ws[4] = (unsigned)iters;
}

// ---------------- Pass 2: only if T < 100, rerun for T steps ----------------
__global__ void __launch_bounds__(BLOCK)
sdf_pass2(const float* __restrict__ dual, const unsigned* __restrict__ ws,
          float* __restrict__ out, int n) {
  const int iters = (int)ws[4];     // uniform s_load
  if (iters >= N_ITERS) return;     // f_100 already written by pass 1

  __shared__ float tile[TILE];
  const int tb  = blockIdx.x * TILE;
  const int tid = threadIdx.x;

  stage_tile(dual, tile, tb, tid, n);
  __syncthreads();

  float2v ndv[EPV], fl[EPV], nb[EPV];
  load_state(tile, tid, ndv, fl, nb);

  for (int t = 0; t < iters; ++t) {
#pragma unroll
    for (int j = 0; j < EPV; ++j) sdf_step_pk(ndv[j], fl[j], nb[j]);
  }

  store_flow(out, fl, tb, tid, n);
}

__global__ void sdf_zero_ws(unsigned* __restrict__ ws) {
  if (threadIdx.x < 8) ws[threadIdx.x] = 0u;
}

extern "C" void kernel_launch(void* const* d_in, const int* in_sizes, int n_in,
                              void* d_out, int out_size, void* d_ws, size_t ws_size,
                              hipStream_t stream) {
  const float* dual = (const float*)d_in[0];
  float* out        = (float*)d_out;
  unsigned* ws      = (unsigned*)d_ws;
  const int n       = in_sizes[0];
  const int nblocks = (n + TILE - 1) / TILE;  // 4096 for n = 16.7M

  sdf_zero_ws<<<1, 32, 0, stream>>>(ws);
  sdf_pass1<<<nblocks, BLOCK, 0, stream>>>(dual, ws, out, n);
  sdf_select_iters<<<1, 1, 0, stream>>>(ws);
  sdf_pass2<<<nblocks, BLOCK, 0, stream>>>(dual, ws, out, n);
}